// FSQ_GumbelSoftmax_1271310319895
// MI455X (gfx1250) — compile-verified
//
#include <hip/hip_runtime.h>
#include <math.h>

// FSQ + hard Gumbel-softmax, fused single pass for MI455X (gfx1250).
// Data path: TDM (tensor_load_to_lds, double-buffered, s_wait_tensorcnt)
//            -> LDS -> V_WMMA_F32_16X16X4_F32 for both skinny GEMMs.
// Bandwidth-bound: ~420 MB HBM traffic -> ~18 us floor @ 23.3 TB/s.

typedef __attribute__((ext_vector_type(2))) float        v2f;
typedef __attribute__((ext_vector_type(8))) float        v8f;
typedef __attribute__((ext_vector_type(4))) unsigned int v4u;
typedef __attribute__((ext_vector_type(8))) int          v8i;
typedef __attribute__((ext_vector_type(4))) int          v4i;

#define M_ROWS          65536          // B*S
#define C_IN            768
#define D_DIM           8
#define L_LVL           8
#define THREADS         256
#define WAVES           8
#define ROWS_PER_BLOCK  128            // 8 waves * 16 rows
#define KC              96             // K-chunk per TDM load (16x96 f32 tile)
#define NCHUNK          (C_IN / KC)    // 8
#define ZROW_PAD        102            // 96 dwords + TDM pad (2 dw / 32 dw)
#define ZBUF_DW         (16 * ZROW_PAD)// 1632 dwords per chunk buffer
#define WC_PITCH        770            // 768 + 2 pad -> conflict-free, even
#define ZQ_N            (M_ROWS * C_IN)
#define INV_CNT         (1.0f / (float)(M_ROWS * D_DIM))

__global__ void fsq_zero_err(float* __restrict__ out) {
  if (threadIdx.x == 0) out[ZQ_N] = 0.0f;
}

// Issue one TDM transfer: z[m0..m0+15, chunk*KC .. +KC) -> LDS (padded rows).
// D# per CDNA5 ISA ch.8: group0 = {count|flags, lds_addr, gaddr_lo, gaddr_hi|type},
// group1 = {flags, dims/strides/tile}, groups 2/3 zero (2D tensor).
__device__ __forceinline__ void tdm_load_chunk(const float* __restrict__ z,
                                               int m0, int chunk,
                                               unsigned lds_byte_addr) {
  unsigned long long ga =
      (unsigned long long)(uintptr_t)(z + (size_t)m0 * C_IN + chunk * KC);
  v4u g0;
  g0.x = 1u;                                             // count=1, user mode
  g0.y = lds_byte_addr;                                  // lds_addr [63:32]
  g0.z = (unsigned)(ga & 0xffffffffull);                 // global_addr lo
  g0.w = (unsigned)((ga >> 32) & 0x1ffffffull) | (2u << 30); // addr hi | type=2
  v8i g1;
  g1[0] = (2 << 16)        // data_size = 4B
        | (1 << 20)        // pad_enable
        | (4 << 22)        // pad_interval: every 32 dwords
        | (1 << 25);       // pad_amount: 2 dwords  -> row stride 102 dw
  g1[1] = (C_IN & 0xffff) << 16;          // tensor_dim0 = 768 (lo16 @ bit48)
  g1[2] = 0;                              // dim0 hi / tensor_dim1 lo16 (65536)
  g1[3] = 1 | (KC << 16);                 // tensor_dim1 hi16=1, tile_dim0=96
  g1[4] = 16;                             // tile_dim1 = 16 rows, tile_dim2=0
  g1[5] = C_IN;                           // tensor_dim0_stride = 768
  g1[6] = 0;
  g1[7] = 0;                              // tensor_dim1_stride unused (2D)
  v4i gz4  = {0, 0, 0, 0};
  v8i gz8  = {0, 0, 0, 0, 0, 0, 0, 0};
  // 6-arg toolchain variant: (g0, g1, g2, g3, g_extra, cpol)
  __builtin_amdgcn_tensor_load_to_lds(g0, g1, gz4, gz4, gz8, 0);
}

__global__ __launch_bounds__(THREADS) void fsq_fused(
    const float* __restrict__ z,  const float* __restrict__ u,
    const float* __restrict__ Wc, const float* __restrict__ bc,
    const float* __restrict__ We, const float* __restrict__ be,
    const float* __restrict__ cb, float* __restrict__ out) {
  __shared__ __align__(16) float sWcT[D_DIM * WC_PITCH];      // WcT 8x770, 24KB
  __shared__ __align__(16) float sWeT[C_IN * D_DIM];          // WeT 768x8, 24KB
  __shared__ __align__(16) float sCb[D_DIM * L_LVL];
  __shared__ __align__(16) float sCodes[WAVES * 16 * D_DIM];  // 4KB
  __shared__ __align__(16) float sZ[WAVES * 2 * ZBUF_DW];     // ~102KB dbl-buf
  __shared__ float sErr;

  const int tid   = threadIdx.x;
  const int wave  = tid >> 5;
  const int lane  = tid & 31;
  const int col   = lane & 15;     // M-row (A) / N-col (B,C) index within 16
  const int khalf = lane >> 4;     // 0 -> K{0,1}, 1 -> K{2,3}

  const int m0 = blockIdx.x * ROWS_PER_BLOCK + wave * 16;
  const unsigned zb0 = (unsigned)(uintptr_t)(&sZ[(wave * 2 + 0) * ZBUF_DW]);
  const unsigned zb1 = (unsigned)(uintptr_t)(&sZ[(wave * 2 + 1) * ZBUF_DW]);

  // Kick off the first two z-tile DMAs; they run while we stage weights.
  tdm_load_chunk(z, m0, 0, zb0);
  tdm_load_chunk(z, m0, 1, zb1);

  // ---- stage transposed weights + codebook into LDS ----
  for (int i = tid; i < C_IN * D_DIM; i += THREADS) {
    sWcT[(i & 7) * WC_PITCH + (i >> 3)] = Wc[i];   // (768x8) -> WcT[d][k]
    const int kk = i / C_IN;
    const int n  = i - kk * C_IN;
    sWeT[n * D_DIM + kk] = We[i];                  // (8x768) -> WeT[n][k]
  }
  if (tid < D_DIM * L_LVL) sCb[tid] = cb[tid];
  if (tid == 0) sErr = 0.0f;
  __syncthreads();

  // =====================================================================
  // Phase 1: compress  zc = z(16x768) * Wc(768x8)  -- 192 WMMA f32 16x16x4
  // Lanes col>=8 read a duplicate Wc column (col&7): dead C cols 8..15.
  // =====================================================================
  v8f c = {};
  const float* wrow  = &sWcT[(col & 7) * WC_PITCH];
  const int    rbase = ZROW_PAD * col + 2 * khalf;     // padded row base in buf
  for (int ch = 0; ch < NCHUNK; ++ch) {
    if (ch == NCHUNK - 1) __builtin_amdgcn_s_wait_tensorcnt(0);
    else                  __builtin_amdgcn_s_wait_tensorcnt(1);
    asm volatile("" ::: "memory");                     // no LDS reads above wait
    const float* buf = &sZ[(wave * 2 + (ch & 1)) * ZBUF_DW];
    const int kg0 = ch * KC + 2 * khalf;
#pragma unroll
    for (int xl = 0; xl < KC; xl += 4) {
      v2f a = *(const v2f*)(buf + rbase + xl + 2 * (xl >> 5)); // ds_load_b64
      v2f b = *(const v2f*)(wrow + kg0 + xl);                  // ds_load_b64
      c = __builtin_amdgcn_wmma_f32_16x16x4_f32(false, a, false, b,
                                                (short)0, c, false, false);
    }
    asm volatile("" ::: "memory");                     // reads done before reuse
    if (ch + 2 < NCHUNK)
      tdm_load_chunk(z, m0, ch + 2, (ch & 1) ? zb1 : zb0);
  }

  // =====================================================================
  // Phase 2: hard Gumbel quantize.  argmax(softmax(x)) == argmax(x).
  // C layout: VGPR r, lane -> zc[m0 + r + 8*khalf, col].
  // =====================================================================
  float qerr = 0.0f;
  if (col < D_DIM) {
    const int d = col;
    const float bcd = bc[d];
#pragma unroll
    for (int r = 0; r < 8; ++r) {
      const int lr  = r + 8 * khalf;
      const int row = m0 + lr;
      const float zcv = c[r] + bcd;
      const float* up = u + (size_t)row * (D_DIM * L_LVL) + d * L_LVL;
      float best = -INFINITY;
      int bi = 0;
#pragma unroll
      for (int l = 0; l < L_LVL; ++l) {
        const float g = -__logf(-__logf(up[l] + 1e-10f));   // gumbel
        const float s = g - fabsf(zcv - sCb[d * L_LVL + l]);
        if (s > best) { best = s; bi = l; }
      }
      const float code = sCb[d * L_LVL + bi];
      const float e = zcv - code;
      qerr += e * e;
      sCodes[(wave * 16 + lr) * D_DIM + d] = code;
    }
  }
  atomicAdd(&sErr, qerr);                                   // ds_add_f32
  __builtin_amdgcn_wave_barrier();   // sCodes writes precede reads below

  // =====================================================================
  // Phase 3: expand  z_q = codes(16x8) * We(8x768) -- 48 tiles x 2 WMMA
  // =====================================================================
  const float* cw = &sCodes[(wave * 16 + col) * D_DIM];
  v2f a0 = *(const v2f*)(cw + 2 * khalf);                   // K 0..3
  v2f a1 = *(const v2f*)(cw + 2 * khalf + 4);               // K 4..7
  for (int n0 = 0; n0 < C_IN; n0 += 16) {
    const float* wn = &sWeT[(n0 + col) * D_DIM + 2 * khalf];
    v2f b0 = *(const v2f*)(wn);                             // ds_load_b64
    v2f b1 = *(const v2f*)(wn + 4);
    v8f dacc = {};
    dacc = __builtin_amdgcn_wmma_f32_16x16x4_f32(false, a0, false, b0,
                                                 (short)0, dacc, false, false);
    dacc = __builtin_amdgcn_wmma_f32_16x16x4_f32(false, a1, false, b1,
                                                 (short)0, dacc, false, false);
    const float bev = be[n0 + col];
#pragma unroll
    for (int r = 0; r < 8; ++r) {
      const int row = m0 + r + 8 * khalf;
      out[(size_t)row * C_IN + n0 + col] = dacc[r] + bev;
    }
  }

  // ---- block-level quantization-error reduction -> one global atomic ----
  __syncthreads();
  if (tid == 0) atomicAdd(out + ZQ_N, sErr * INV_CNT);
}

extern "C" void kernel_launch(void* const* d_in, const int* in_sizes, int n_in,
                              void* d_out, int out_size, void* d_ws,
                              size_t ws_size, hipStream_t stream) {
  (void)in_sizes; (void)n_in; (void)out_size; (void)d_ws; (void)ws_size;
  const float* z  = (const float*)d_in[0];
  const float* u  = (const float*)d_in[1];
  const float* Wc = (const float*)d_in[2];
  const float* bc = (const float*)d_in[3];
  const float* We = (const float*)d_in[4];
  const float* be = (const float*)d_in[5];
  const float* cb = (const float*)d_in[6];
  // d_in[7] = codebook_mask: all levels == L -> mask all-true, folded away.
  float* out = (float*)d_out;

  fsq_zero_err<<<1, 1, 0, stream>>>(out);
  fsq_fused<<<M_ROWS / ROWS_PER_BLOCK, THREADS, 0, stream>>>(
      z, u, Wc, bc, We, be, cb, out);
}